// ADJLayer_46050639347845
// MI455X (gfx1250) — compile-verified
//
#include <hip/hip_runtime.h>
#include <hip/hip_bf16.h>

// GAT-style attention layer for MI455X (gfx1250), wave32 + WMMA f32 16x16x4.
//
// Passes:
//  k1: read adj (512MB) -> colsum[128] + Gram G = X^T X [128x128] via WMMA
//  k2: fold BN1/BN2 analytically (var2 from Gram) -> W''[128x64], b''[64]
//  k3: read adj (512MB) -> z = x@W''+b'' via WMMA, leaky, dot a -> s[1M]
//      (stored in d_out), accumulate sum(s), sum(s^2) for BN3
//  k4: BN3 + leaky + mask + row softmax, in-place on d_out
//
// All WMMA fragments are single aligned 8-byte LDS loads (ds_load_b64) so the
// v2f operands land directly in even VGPR pairs with no packing movs.

typedef __attribute__((ext_vector_type(2))) float v2f;
typedef __attribute__((ext_vector_type(8))) float v8f;

#define RTOT (1024 * 1024)
#define FIN 128
#define FH 64
#define EPSV 1e-5f
#define SLOPE 0.2f
#define NEGV -9e15f

__device__ __forceinline__ v8f wmma_f32_4(v2f a, v2f b, v8f c) {
  // (neg_a, A, neg_b, B, c_mod, C, reuse_a, reuse_b)
  return __builtin_amdgcn_wmma_f32_16x16x4_f32(false, a, false, b, (short)0, c,
                                               false, false);
}

// ---------------------------------------------------------------------------
// Kernel 1: column sums + Gram matrix G = X^T X over 1M rows.
// 256 WGs x 256 threads (8 waves). Each WG: 4096 rows, staged 16 rows/iter.
// LDS tile is stored TRANSPOSED (ldsT[col*18 + row]) so that the K-pairs of
// both A and B fragments are contiguous -> one ds_load_b64 per fragment.
// Wave w owns Gram row-block i=w (16 rows), loops 8 column blocks.
// ---------------------------------------------------------------------------
__global__ __launch_bounds__(256) void k1_stats_gram(
    const float* __restrict__ adj, float* __restrict__ colsum,
    float* __restrict__ gram) {
  __shared__ __align__(16) float ldsT[128 * 18];  // [col][row], stride 18
  __shared__ float red4[256 * 4];
  const int tid = threadIdx.x;
  const int lane = tid & 31;
  const int wave = tid >> 5;  // i-block 0..7
  const int m = lane & 15;
  const int h = lane >> 4;
  const long r0 = (long)blockIdx.x * 4096;

  v8f acc[8];
#pragma unroll
  for (int jb = 0; jb < 8; ++jb)
#pragma unroll
    for (int e = 0; e < 8; ++e) acc[jb][e] = 0.0f;

  float csum[4] = {0.0f, 0.0f, 0.0f, 0.0f};

  for (int it = 0; it < 256; ++it) {  // 256 * 16 = 4096 rows
    __syncthreads();
    const float4* src4 = (const float4*)(adj + (r0 + (long)it * 16) * FIN);
#pragma unroll
    for (int j = 0; j < 2; ++j) {
      int e4 = tid + j * 256;  // 0..511 float4s (16 rows x 32 float4)
      int row = e4 >> 5;
      int c0 = (e4 & 31) * 4;  // == 4*(tid&31) for both j
      float4 v = src4[e4];
      ldsT[(c0 + 0) * 18 + row] = v.x;
      ldsT[(c0 + 1) * 18 + row] = v.y;
      ldsT[(c0 + 2) * 18 + row] = v.z;
      ldsT[(c0 + 3) * 18 + row] = v.w;
      csum[0] += v.x;
      csum[1] += v.y;
      csum[2] += v.z;
      csum[3] += v.w;
    }
    __syncthreads();
#pragma unroll
    for (int ks = 0; ks < 4; ++ks) {  // 4 K-steps of 4 rows
      const int ka = ks * 4 + 2 * h;  // even -> 8B-aligned v2f load
      v2f af = *(const v2f*)&ldsT[(wave * 16 + m) * 18 + ka];
#pragma unroll
      for (int jb = 0; jb < 8; ++jb) {
        v2f bf = *(const v2f*)&ldsT[(jb * 16 + m) * 18 + ka];
        acc[jb] = wmma_f32_4(af, bf, acc[jb]);
      }
    }
  }
  // C/D layout: VGPR v -> rows v (lanes0-15) and v+8 (lanes16-31), col = m
#pragma unroll
  for (int jb = 0; jb < 8; ++jb)
#pragma unroll
    for (int v = 0; v < 8; ++v) {
      int gi = wave * 16 + v + 8 * h;
      int gj = jb * 16 + m;
      atomicAdd(&gram[gi * FIN + gj], acc[jb][v]);
    }
  // column sums: thread t holds partials for cols 4*(t&31)+q
  red4[tid * 4 + 0] = csum[0];
  red4[tid * 4 + 1] = csum[1];
  red4[tid * 4 + 2] = csum[2];
  red4[tid * 4 + 3] = csum[3];
  __syncthreads();
  if (tid < 128) {  // col c = tid ; contributors: t with (t&31) == c>>2
    int base = tid >> 2;
    int q = tid & 3;
    float s = 0.0f;
#pragma unroll
    for (int r = 0; r < 8; ++r) s += red4[(base + r * 32) * 4 + q];
    atomicAdd(&colsum[tid], s);
  }
}

// ---------------------------------------------------------------------------
// Kernel 2 (1 WG): fold BN1 into W -> W', derive mu2/var2 from Gram, fold
// BN2 -> W''[128x64], b''[64].
// ---------------------------------------------------------------------------
__global__ __launch_bounds__(256) void k2_prepare(
    const float* __restrict__ colsum, const float* __restrict__ gram,
    const float* __restrict__ W, const float* __restrict__ g1,
    const float* __restrict__ b1, const float* __restrict__ g2,
    const float* __restrict__ b2, float* __restrict__ wpp,
    float* __restrict__ bpp) {
  __shared__ float A1[128], B1[128], MU1[128];
  __shared__ float Wp[128 * 64];
  __shared__ float bp[64], mu2l[64], A2[64];
  __shared__ float qred[256];
  const int tid = threadIdx.x;
  const float invR = 1.0f / (float)RTOT;

  if (tid < 128) {
    float mu = colsum[tid] * invR;
    float var = gram[tid * 128 + tid] * invR - mu * mu;
    float s = rsqrtf(var + EPSV);
    float a1 = g1[tid] * s;
    A1[tid] = a1;
    B1[tid] = b1[tid] - mu * a1;
    MU1[tid] = mu;
  }
  __syncthreads();
  for (int e = tid; e < 128 * 64; e += 256) Wp[e] = A1[e >> 6] * W[e];
  __syncthreads();
  if (tid < 64) {
    float bpj = 0.0f, mj = 0.0f;
    for (int i = 0; i < 128; ++i) {
      bpj += B1[i] * W[i * 64 + tid];
      mj += MU1[i] * Wp[i * 64 + tid];
    }
    bp[tid] = bpj;
    mu2l[tid] = mj + bpj;  // mu2
  }
  __syncthreads();
  {  // q_j = w'_j^T G w'_j ; 4 partial-sums per j over i-ranges
    int j = tid & 63, part = tid >> 6;
    float q = 0.0f;
    for (int i = part * 32; i < part * 32 + 32; ++i) {
      const float* grow = &gram[i * 128];
      float gi = 0.0f;
      for (int k = 0; k < 128; ++k) gi += grow[k] * Wp[k * 64 + j];
      q += Wp[i * 64 + j] * gi;
    }
    qred[tid] = q;
  }
  __syncthreads();
  if (tid < 64) {
    float q = qred[tid] + qred[tid + 64] + qred[tid + 128] + qred[tid + 192];
    float mj = mu2l[tid] - bp[tid];
    float var2 = q * invR - mj * mj;  // E[(x.w')^2] - E[x.w']^2
    float a2 = g2[tid] * rsqrtf(var2 + EPSV);
    A2[tid] = a2;
    bpp[tid] = bp[tid] * a2 + (b2[tid] - mu2l[tid] * a2);
  }
  __syncthreads();
  for (int e = tid; e < 128 * 64; e += 256) wpp[e] = Wp[e] * A2[e & 63];
}

// ---------------------------------------------------------------------------
// Kernel 3: main pass. 1024 WGs x 256 threads; each WG does 8 blocks of
// 128 rows. Wave w computes rows [16w,16w+16): z = x @ W'' + b'' via WMMA
// (4 tiles of 16x16, K=128), then s = sum_j leaky(z_j)*a_j per row.
// x tile is row-major (A K-pairs contiguous); W'' is stored transposed in
// LDS (B K-pairs contiguous) -> every fragment is one ds_load_b64.
// ---------------------------------------------------------------------------
__global__ __launch_bounds__(256) void k3_main(
    const float* __restrict__ adj, const float* __restrict__ wpp,
    const float* __restrict__ bpp, const float* __restrict__ avec,
    float* __restrict__ s_out, float* __restrict__ gsum) {
  __shared__ __align__(16) float xl[128 * 132];   // [row][col], stride 132
  __shared__ __align__(16) float wlT[64 * 130];   // [n][k], stride 130
  __shared__ float bl[64], al[64];
  const int tid = threadIdx.x;
  const int lane = tid & 31;
  const int wave = tid >> 5;
  const int m = lane & 15;
  const int h = lane >> 4;

  for (int e = tid; e < 128 * 64; e += 256)
    wlT[(e & 63) * 130 + (e >> 6)] = wpp[e];
  if (tid < 64) {
    bl[tid] = bpp[tid];
    al[tid] = avec[tid];
  }

  float ssum = 0.0f, ssq = 0.0f;

  for (int blk = 0; blk < 8; ++blk) {
    long rbase = ((long)blockIdx.x * 8 + blk) * 128;
    __syncthreads();
    const float4* src = (const float4*)(adj + rbase * FIN);
#pragma unroll
    for (int j = 0; j < 16; ++j) {
      int e4 = tid + j * 256;  // 0..4095 float4s
      int row = e4 >> 5;
      int c4 = e4 & 31;
      float4 v = src[e4];
      float* dst = &xl[row * 132 + c4 * 4];
      dst[0] = v.x;
      dst[1] = v.y;
      dst[2] = v.z;
      dst[3] = v.w;
    }
    __syncthreads();

    const float* xr = &xl[(wave * 16 + m) * 132];
    v8f acc[4];
#pragma unroll
    for (int t = 0; t < 4; ++t)
#pragma unroll
      for (int e = 0; e < 8; ++e) acc[t][e] = 0.0f;

#pragma unroll 4
    for (int ks = 0; ks < 32; ++ks) {
      const int ka = ks * 4 + 2 * h;  // even -> aligned v2f
      v2f af = *(const v2f*)&xr[ka];  // A[m][k] = x[row0+m][k]
#pragma unroll
      for (int jb = 0; jb < 4; ++jb) {
        // B[k][n] = W''[k][16jb+n] ; wlT[n*130 + k]
        v2f bf = *(const v2f*)&wlT[(jb * 16 + m) * 130 + ka];
        acc[jb] = wmma_f32_4(af, bf, acc[jb]);
      }
    }
    // epilogue: per row r, s_r = sum_j leaky(z_rj + b''_j) * a_j
#pragma unroll
    for (int v = 0; v < 8; ++v) {
      float part = 0.0f;
#pragma unroll
      for (int jb = 0; jb < 4; ++jb) {
        int col = jb * 16 + m;
        float z = acc[jb][v] + bl[col];
        z = (z >= 0.0f) ? z : SLOPE * z;
        part += z * al[col];
      }
      // reduce across the 16 lanes of each half-wave
      part += __shfl_xor(part, 1);
      part += __shfl_xor(part, 2);
      part += __shfl_xor(part, 4);
      part += __shfl_xor(part, 8);
      if (m == 0) {
        int row = wave * 16 + v + 8 * h;
        s_out[rbase + row] = part;
        ssum += part;
        ssq += part * part;
      }
    }
  }
  if (m == 0) {  // lanes 0 and 16 of each wave hold valid sums
    atomicAdd(&gsum[0], ssum);
    atomicAdd(&gsum[1], ssq);
  }
}

// ---------------------------------------------------------------------------
// Kernel 4: BN3 + leaky + mask + row softmax, in-place on d_out.
// One WG (256 threads) per row of 1024.
// ---------------------------------------------------------------------------
__global__ __launch_bounds__(256) void k4_softmax(
    const float* __restrict__ adj_mean, const float* __restrict__ gsum,
    const float* __restrict__ g3, const float* __restrict__ b3,
    float* __restrict__ out) {
  __shared__ float red[256];
  const int row = blockIdx.x, tid = threadIdx.x;
  const float invR = 1.0f / (float)RTOT;
  float mu = gsum[0] * invR;
  float var = gsum[1] * invR - mu * mu;
  float a3 = g3[0] * rsqrtf(var + EPSV);
  float b3v = b3[0] - mu * a3;

  float vals[4];
  float mx = -3.4e38f;
#pragma unroll
  for (int j = 0; j < 4; ++j) {
    int c = tid + j * 256;
    float e = out[row * 1024 + c] * a3 + b3v;
    e = (e >= 0.0f) ? e : SLOPE * e;
    float att = (adj_mean[row * 1024 + c] > 0.0f) ? e : NEGV;
    vals[j] = att;
    mx = fmaxf(mx, att);
  }
  red[tid] = mx;
  __syncthreads();
  for (int off = 128; off > 0; off >>= 1) {
    if (tid < off) red[tid] = fmaxf(red[tid], red[tid + off]);
    __syncthreads();
  }
  mx = red[0];
  __syncthreads();
  float sum = 0.0f;
#pragma unroll
  for (int j = 0; j < 4; ++j) {
    vals[j] = expf(vals[j] - mx);
    sum += vals[j];
  }
  red[tid] = sum;
  __syncthreads();
  for (int off = 128; off > 0; off >>= 1) {
    if (tid < off) red[tid] += red[tid + off];
    __syncthreads();
  }
  float inv = 1.0f / red[0];
#pragma unroll
  for (int j = 0; j < 4; ++j) out[row * 1024 + tid + j * 256] = vals[j] * inv;
}

extern "C" void kernel_launch(void* const* d_in, const int* in_sizes, int n_in,
                              void* d_out, int out_size, void* d_ws,
                              size_t ws_size, hipStream_t stream) {
  const float* adj = (const float*)d_in[0];
  const float* adj_mean = (const float*)d_in[1];
  const float* W = (const float*)d_in[2];
  const float* a = (const float*)d_in[3];
  const float* g1 = (const float*)d_in[4];
  const float* b1 = (const float*)d_in[5];
  const float* g2 = (const float*)d_in[6];
  const float* b2 = (const float*)d_in[7];
  const float* g3 = (const float*)d_in[8];
  const float* b3 = (const float*)d_in[9];
  float* out = (float*)d_out;

  float* ws = (float*)d_ws;
  float* gsum = ws;                   // 2 floats  (Sum s, Sum s^2)
  float* colsum = ws + 16;            // 128 floats
  float* gram = ws + 256;             // 128*128 floats
  float* wpp = ws + 256 + 16384;      // 128*64 floats
  float* bpp = wpp + 128 * 64;        // 64 floats

  // zero the accumulators (gsum + colsum + gram) every call
  hipMemsetAsync(d_ws, 0, (size_t)(256 + 16384) * sizeof(float), stream);

  k1_stats_gram<<<256, 256, 0, stream>>>(adj, colsum, gram);
  k2_prepare<<<1, 256, 0, stream>>>(colsum, gram, W, g1, b1, g2, b2, wpp, bpp);
  k3_main<<<1024, 256, 0, stream>>>(adj, wpp, bpp, a, out, gsum);
  k4_softmax<<<1024, 256, 0, stream>>>(adj_mean, gsum, g3, b3, out);
}